// HGNN_51986284151155
// MI455X (gfx1250) — compile-verified
//
#include <hip/hip_runtime.h>
#include <hip/hip_bf16.h>

typedef float v2f __attribute__((ext_vector_type(2)));
typedef float v8f __attribute__((ext_vector_type(8)));

#define H 128
#define FIN 64
#define KP 68        // 67 padded to 68 (17 k-steps of 4)
#define AP 132       // phase-3 A stride (128+4, conflict-free)
#define EPSF 1e-5f

// order-preserving float -> uint key (monotone for all finite floats)
__device__ __forceinline__ unsigned fkey(float f) {
    unsigned b = __float_as_uint(f);
    return (b & 0x80000000u) ? ~b : (b | 0x80000000u);
}
__device__ __forceinline__ float funkey(unsigned k) {
    unsigned b = (k & 0x80000000u) ? (k & 0x7fffffffu) : ~k;
    return __uint_as_float(b);
}

__global__ void k_zero(unsigned* __restrict__ p, int total) {
    int i = blockIdx.x * blockDim.x + threadIdx.x;
    if (i < total) p[i] = 0u;
}

// Phase 1: fused gather + GEMM(E x 67 @ 67 x 128) + bias + relu
//          + column sum/sumsq (for BN1) + atomic scatter-max of sign(gamma)*h
__global__ __launch_bounds__(256)
void k_edge_gemm(const float* __restrict__ last_coors,
                 const float* __restrict__ last_features,
                 const float* __restrict__ current_coors,
                 const int*   __restrict__ edge, int E,
                 const float* __restrict__ W_in,
                 const float* __restrict__ b_in,
                 const float* __restrict__ gamma_in,
                 unsigned* __restrict__ segkey,
                 double* __restrict__ gsum, double* __restrict__ gsumsq)
{
    __shared__ __align__(16) float X[128 * KP];   // 34816 B
    __shared__ __align__(16) float Wl[KP * H];    // 34816 B
    __shared__ int   curL[128];
    __shared__ float csum[H];
    __shared__ float csq[H];

    const int tid = threadIdx.x;
    const int e0  = blockIdx.x * 128;

    if (tid < H) { csum[tid] = 0.f; csq[tid] = 0.f; }

    // stage W_in (67x128 row-major) into LDS, row 67 zero-padded
    for (int j = tid; j < (KP * H) / 4; j += 256) {
        int k  = j >> 5;            // 32 float4 per row
        int n4 = (j & 31) * 4;
        float4 w = make_float4(0.f, 0.f, 0.f, 0.f);
        if (k < FIN + 3) w = *(const float4*)(W_in + (size_t)k * H + n4);
        *(float4*)(Wl + k * H + n4) = w;
    }
    // stage 128 gathered rows of x = [feat(64), dcoor(3), 0]
    for (int j = tid; j < 128 * 17; j += 256) {
        int le = j / 17;
        int ch = j % 17;
        int e  = e0 + le;
        float4 xv = make_float4(0.f, 0.f, 0.f, 0.f);
        if (e < E) {
            int li = edge[E + e];
            if (ch < 16) {
                xv = *(const float4*)(last_features + (size_t)li * FIN + ch * 4);
            } else {
                int ci = edge[e];
                xv.x = last_coors[li * 3 + 0] - current_coors[ci * 3 + 0];
                xv.y = last_coors[li * 3 + 1] - current_coors[ci * 3 + 1];
                xv.z = last_coors[li * 3 + 2] - current_coors[ci * 3 + 2];
            }
        }
        *(float4*)(X + le * KP + ch * 4) = xv;
        if (ch == 0) curL[le] = (e < E) ? edge[e] : 0;
    }
    __syncthreads();

    const int lane = tid & 31;
    const int wv   = tid >> 5;      // wave 0..7 -> rows 16w..16w+15
    const int g    = lane >> 4;     // half-wave
    const int lm   = lane & 15;
    const int rowBase = wv * 16;

    for (int nt = 0; nt < 8; ++nt) {
        const int n = nt * 16 + lm;
        v8f acc = {};
        for (int ks = 0; ks < 17; ++ks) {
            const int kk = ks * 4 + g * 2;          // A/B element k = v + 2g
            v2f a = *(const v2f*)(X + (rowBase + lm) * KP + kk);
            v2f b;
            b.x = Wl[kk * H + n];
            b.y = Wl[(kk + 1) * H + n];
            acc = __builtin_amdgcn_wmma_f32_16x16x4_f32(
                      false, a, false, b, (short)0, acc, false, false);
        }
        const float gam  = gamma_in[n];
        const float bias = b_in[n];
        float lsum = 0.f, lsq = 0.f;
#pragma unroll
        for (int v = 0; v < 8; ++v) {
            const int le = rowBase + v + 8 * g;     // D row M = v + 8g
            const int e  = e0 + le;
            if (e < E) {
                float h = acc[v] + bias;
                h = h > 0.f ? h : 0.f;
                lsum += h; lsq += h * h;
                const float t = (gam >= 0.f) ? h : -h;
                atomicMax(segkey + (size_t)curL[le] * H + n, fkey(t));
            }
        }
        atomicAdd(&csum[n], lsum);
        atomicAdd(&csq[n],  lsq);
    }
    __syncthreads();
    if (tid < H) {
        atomicAdd(gsum   + tid, (double)csum[tid]);
        atomicAdd(gsumsq + tid, (double)csq[tid]);
    }
}

// BN affine params from accumulated stats: a = gamma*rsqrt(var+eps), c = beta - a*mu
__global__ void k_bnparams(const double* __restrict__ gsum,
                           const double* __restrict__ gsumsq, int rows,
                           const float* __restrict__ gamma,
                           const float* __restrict__ beta,
                           float* __restrict__ aOut, float* __restrict__ cOut,
                           int absFlag)
{
    int n = threadIdx.x;
    double mu  = gsum[n] / (double)rows;
    double var = gsumsq[n] / (double)rows - mu * mu;
    if (var < 0.0) var = 0.0;
    float a = gamma[n] * (float)(1.0 / sqrt(var + (double)EPSF));
    float c = beta[n] - a * (float)mu;
    aOut[n] = absFlag ? fabsf(a) : a;
    cOut[n] = c;
}

// Phase 2b: decode per-(segment,col) key -> agg = max(0, |a|*t_ext + c); empty -> 0
__global__ void k_decode(unsigned* __restrict__ seg,
                         const float* __restrict__ absA,
                         const float* __restrict__ cvec, int total)
{
    int i = blockIdx.x * blockDim.x + threadIdx.x;
    if (i >= total) return;
    unsigned k = seg[i];
    int n = i & (H - 1);
    float out = 0.f;
    if (k != 0u) {
        float v = absA[n] * funkey(k) + cvec[n];
        out = v > 0.f ? v : 0.f;
    }
    ((float*)seg)[i] = out;   // in-place reuse as fp32 agg
}

// Phase 3: agg(M x 128) @ W_out(128 x 128) + bias + relu -> d_out, + BN2 stats
__global__ __launch_bounds__(256)
void k_gemm2(const float* __restrict__ agg, int M,
             const float* __restrict__ W_out,
             const float* __restrict__ b_out,
             float* __restrict__ h2,
             double* __restrict__ gsum, double* __restrict__ gsumsq)
{
    __shared__ __align__(16) float A[128 * AP];   // 67584 B
    __shared__ __align__(16) float Wl[128 * 128]; // 65536 B
    __shared__ float csum[H];
    __shared__ float csq[H];

    const int tid = threadIdx.x;
    const int r0  = blockIdx.x * 128;

    if (tid < H) { csum[tid] = 0.f; csq[tid] = 0.f; }

    for (int j = tid; j < 128 * 32; j += 256) {
        int r  = j >> 5;
        int n4 = (j & 31) * 4;
        float4 av = make_float4(0.f, 0.f, 0.f, 0.f);
        if (r0 + r < M) av = *(const float4*)(agg + (size_t)(r0 + r) * H + n4);
        *(float4*)(A + r * AP + n4) = av;
        *(float4*)(Wl + r * H + n4) = *(const float4*)(W_out + (size_t)r * H + n4);
    }
    __syncthreads();

    const int lane = tid & 31;
    const int wv   = tid >> 5;
    const int g    = lane >> 4;
    const int lm   = lane & 15;
    const int rowBase = wv * 16;

    for (int nt = 0; nt < 8; ++nt) {
        const int n = nt * 16 + lm;
        v8f acc = {};
        for (int ks = 0; ks < 32; ++ks) {
            const int kk = ks * 4 + g * 2;
            v2f a = *(const v2f*)(A + (rowBase + lm) * AP + kk);
            v2f b;
            b.x = Wl[kk * H + n];
            b.y = Wl[(kk + 1) * H + n];
            acc = __builtin_amdgcn_wmma_f32_16x16x4_f32(
                      false, a, false, b, (short)0, acc, false, false);
        }
        const float bias = b_out[n];
        float lsum = 0.f, lsq = 0.f;
#pragma unroll
        for (int v = 0; v < 8; ++v) {
            const int r = r0 + rowBase + v + 8 * g;
            if (r < M) {
                float h = acc[v] + bias;
                h = h > 0.f ? h : 0.f;
                lsum += h; lsq += h * h;
                h2[(size_t)r * H + n] = h;
            }
        }
        atomicAdd(&csum[n], lsum);
        atomicAdd(&csq[n],  lsq);
    }
    __syncthreads();
    if (tid < H) {
        atomicAdd(gsum   + tid, (double)csum[tid]);
        atomicAdd(gsumsq + tid, (double)csq[tid]);
    }
}

// Phase 5: in-place BN apply on d_out
__global__ void k_bn_apply(float* __restrict__ h2,
                           const float* __restrict__ a2,
                           const float* __restrict__ c2, int total)
{
    int i = blockIdx.x * blockDim.x + threadIdx.x;
    if (i >= total) return;
    int n = i & (H - 1);
    h2[i] = a2[n] * h2[i] + c2[n];
}

extern "C" void kernel_launch(void* const* d_in, const int* in_sizes, int n_in,
                              void* d_out, int out_size, void* d_ws, size_t ws_size,
                              hipStream_t stream) {
    const float* last_coors    = (const float*)d_in[0];
    const float* last_features = (const float*)d_in[1];
    const float* current_coors = (const float*)d_in[2];
    const int*   edge          = (const int*)d_in[3];
    const float* W_in     = (const float*)d_in[4];
    const float* b_in     = (const float*)d_in[5];
    const float* gamma_in = (const float*)d_in[6];
    const float* beta_in  = (const float*)d_in[7];
    const float* W_out    = (const float*)d_in[8];
    const float* b_out    = (const float*)d_in[9];
    const float* gamma_out = (const float*)d_in[10];
    const float* beta_out  = (const float*)d_in[11];

    const int E = in_sizes[3] / 2;
    const int M = in_sizes[2] / 3;

    char* ws = (char*)d_ws;
    double* sum1   = (double*)(ws + 0);
    double* sumsq1 = (double*)(ws + 1024);
    double* sum2   = (double*)(ws + 2048);
    double* sumsq2 = (double*)(ws + 3072);
    float*  absA   = (float*)(ws + 4096);
    float*  cvec   = (float*)(ws + 4608);
    float*  a2     = (float*)(ws + 5120);
    float*  c2     = (float*)(ws + 5632);
    unsigned* segkey = (unsigned*)(ws + 8192);
    float* out = (float*)d_out;

    // zero: stat accumulators (first 8192 B) + segkey region (contiguous)
    const int ztot = 2048 + M * H;
    k_zero<<<(ztot + 255) / 256, 256, 0, stream>>>((unsigned*)ws, ztot);

    k_edge_gemm<<<(E + 127) / 128, 256, 0, stream>>>(
        last_coors, last_features, current_coors, edge, E,
        W_in, b_in, gamma_in, segkey, sum1, sumsq1);

    k_bnparams<<<1, 128, 0, stream>>>(sum1, sumsq1, E, gamma_in, beta_in,
                                      absA, cvec, 1);

    k_decode<<<(M * H + 255) / 256, 256, 0, stream>>>(segkey, absA, cvec, M * H);

    k_gemm2<<<(M + 127) / 128, 256, 0, stream>>>(
        (const float*)segkey, M, W_out, b_out, out, sum2, sumsq2);

    k_bnparams<<<1, 128, 0, stream>>>(sum2, sumsq2, M, gamma_out, beta_out,
                                      a2, c2, 0);

    k_bn_apply<<<(M * H + 255) / 256, 256, 0, stream>>>(out, a2, c2, M * H);
}